// Idx2PixelBCLayer_7541962572383
// MI455X (gfx1250) — compile-verified
//
#include <hip/hip_runtime.h>
#include <hip/hip_bf16.h>

// Bicubic (Catmull-Rom) gather: 2048x2048x8 f32 image, 2M points -> [2M, 8] f32.
// Memory-gather bound; image (128 MB) fits MI455X 192 MB L2 -> keep image RT-cached,
// stream coords/out with NT hints. 2 lanes (wave32 pair) per point, b128 tap loads.

#define IMG_H 2048
#define IMG_W 2048
#define IMG_C 8

typedef float v4f __attribute__((ext_vector_type(4)));
typedef float v2f __attribute__((ext_vector_type(2)));

__device__ __forceinline__ void cr_weights(float t, float w[4]) {
  // w0 = -0.5 t^3 +     t^2 - 0.5 t
  // w1 =  1.5 t^3 - 2.5 t^2 + 1
  // w2 = -1.5 t^3 + 2.0 t^2 + 0.5 t
  // w3 =  0.5 t^3 - 0.5 t^2
  float t2 = t * t;
  float t3 = t2 * t;
  w[0] = fmaf(-0.5f, t3, fmaf(1.0f, t2, -0.5f * t));
  w[1] = fmaf(1.5f, t3, fmaf(-2.5f, t2, 1.0f));
  w[2] = fmaf(-1.5f, t3, fmaf(2.0f, t2, 0.5f * t));
  w[3] = fmaf(0.5f, t3, -0.5f * t2);
}

__global__ __launch_bounds__(256) void bicubic_gather_kernel(
    const float* __restrict__ coords,   // [N,2] f32: axis0 = row (H), axis1 = col (W)
    const float* __restrict__ img,      // [H,W,C] f32
    float* __restrict__ out,            // [N,C] f32
    int n_points) {
  const int gid  = blockIdx.x * blockDim.x + threadIdx.x;
  const int p    = gid >> 1;        // point index
  const int half = gid & 1;         // which 4 of the 8 channels this lane owns
  if (p >= n_points) return;

  // Prefetch the coords stream well ahead of this wave (gfx1250 global_prefetch_b8).
  __builtin_prefetch(coords + 2 * (size_t)p + 4096, 0, 1);

  // Streaming read of this point's coords (non-temporal: don't pollute L2,
  // which we want to keep holding the 128 MB image).
  v2f cd = __builtin_nontemporal_load((const v2f*)(coords + 2 * (size_t)p));

  // Clamp to [1, dim-3] as in the reference, then split into base + frac.
  float cy = fminf(fmaxf(cd.x, 1.0f), (float)(IMG_H - 3));  // axis 0 -> rows
  float cx = fminf(fmaxf(cd.y, 1.0f), (float)(IMG_W - 3));  // axis 1 -> cols
  float fy = floorf(cy);
  float fx = floorf(cx);
  const int by = (int)fy;
  const int bx = (int)fx;
  const float ty = cy - fy;
  const float tx = cx - fx;

  float wi[4];  // weights along axis 0 (rows,   i taps)
  float wj[4];  // weights along axis 1 (cols,   j taps)
  cr_weights(ty, wi);
  cr_weights(tx, wj);

  // Base of the 4x4 footprint; this lane reads channels [half*4, half*4+4).
  const float* base =
      img + ((size_t)(by - 1) * IMG_W + (size_t)(bx - 1)) * IMG_C + half * 4;

  // Issue all 16 b128 tap loads before consuming any (maximize outstanding
  // LOADcnt for L2 latency hiding). Lane pairs cover contiguous 32-B segments.
  v4f tap[4][4];
#pragma unroll
  for (int i = 0; i < 4; ++i) {
    const float* row = base + (size_t)i * (IMG_W * IMG_C);
#pragma unroll
    for (int j = 0; j < 4; ++j) {
      tap[i][j] = *(const v4f*)(row + j * IMG_C);
    }
  }

  v4f acc = (v4f)(0.0f);
#pragma unroll
  for (int i = 0; i < 4; ++i) {
#pragma unroll
    for (int j = 0; j < 4; ++j) {
      const float w = wi[i] * wj[j];
      acc += tap[i][j] * w;   // vector FMAs (v_fma_f32 / v_pk_fma_f32)
    }
  }

  // Streaming, coalesced b128 store with NT hint (written once, never re-read).
  __builtin_nontemporal_store(acc, (v4f*)(out + (size_t)p * IMG_C + half * 4));
}

extern "C" void kernel_launch(void* const* d_in, const int* in_sizes, int n_in,
                              void* d_out, int out_size, void* d_ws, size_t ws_size,
                              hipStream_t stream) {
  (void)n_in; (void)d_ws; (void)ws_size; (void)out_size;
  const float* coords = (const float*)d_in[0];  // [N,2] f32
  const float* img    = (const float*)d_in[1];  // [2048,2048,8] f32
  float* out          = (float*)d_out;          // [N,8] f32

  const int n_points = in_sizes[0] / 2;         // coords has 2N floats
  const int threads  = 2 * n_points;            // 2 lanes per point
  const int block    = 256;                     // 8 wave32s per block
  const int grid     = (threads + block - 1) / block;

  bicubic_gather_kernel<<<grid, block, 0, stream>>>(coords, img, out, n_points);
}